// SamplesLoss_8272107012155
// MI455X (gfx1250) — compile-verified
//
#include <hip/hip_runtime.h>
#include <math.h>

// ---------------------------------------------------------------------------
// Sinkhorn OT (balanced, eps=0.25, 20 iters + final extrapolation)
// N = M = 4096, D = 128.  P[i][j] = -C[i][j]/eps  kept resident in L2 (64MB).
// GEMM via V_WMMA_F32_16X16X4_F32 (fp32 tensor path for precision).
// ---------------------------------------------------------------------------

typedef float v2f __attribute__((ext_vector_type(2)));
typedef float v8f __attribute__((ext_vector_type(8)));

#define NPT   4096
#define DDIM  128
#define EPSR  0.25f
#define IEPS  4.0f
#define NIT   20

// -------------------- weight normalization + init --------------------------
__global__ void prep_weights(const float* __restrict__ a, const float* __restrict__ b,
                             float* __restrict__ a_n, float* __restrict__ b_n,
                             float* __restrict__ log_a, float* __restrict__ log_b,
                             float* __restrict__ f, float* __restrict__ g) {
    __shared__ float red[1024];
    const int t = threadIdx.x;

    float s = 0.f;
    for (int i = t; i < NPT; i += 1024) s += a[i];
    red[t] = s; __syncthreads();
    for (int off = 512; off > 0; off >>= 1) {
        if (t < off) red[t] += red[t + off];
        __syncthreads();
    }
    const float za = fmaxf(red[0], 1e-40f);
    __syncthreads();

    s = 0.f;
    for (int i = t; i < NPT; i += 1024) s += b[i];
    red[t] = s; __syncthreads();
    for (int off = 512; off > 0; off >>= 1) {
        if (t < off) red[t] += red[t + off];
        __syncthreads();
    }
    const float zb = fmaxf(red[0], 1e-40f);

    for (int i = t; i < NPT; i += 1024) {
        float an = a[i] / za;
        float bn = b[i] / zb;
        a_n[i] = an;  log_a[i] = __logf(an);
        b_n[i] = bn;  log_b[i] = __logf(bn);
        f[i] = 0.f;   g[i] = 0.f;
    }
}

// -------------------- per-row squared norms --------------------------------
__global__ void row_sq(const float* __restrict__ x, float* __restrict__ out) {
    const int lane = threadIdx.x & 31;
    const int row  = blockIdx.x * (blockDim.x >> 5) + (threadIdx.x >> 5);
    const float* xr = x + (size_t)row * DDIM;
    float s = 0.f;
    #pragma unroll
    for (int k = 0; k < DDIM; k += 32) {
        float v = xr[k + lane];
        s += v * v;
    }
    #pragma unroll
    for (int off = 16; off > 0; off >>= 1) s += __shfl_xor(s, off, 32);
    if (lane == 0) out[row] = s;
}

// -------------------- cost GEMM: P = 4*(2*x.yT - xsq - ysq) ----------------
// One wave computes a 16 x 64 tile (1 A-frag reused over 4 B-frags).
#define NTJ 4
__global__ void gemm_cost(const float* __restrict__ x, const float* __restrict__ y,
                          const float* __restrict__ xsq, const float* __restrict__ ysq,
                          float* __restrict__ P) {
    const int lane = threadIdx.x & 31;
    const int wave = blockIdx.x * (blockDim.x >> 5) + (threadIdx.x >> 5);
    const int tilesJ = NPT / (16 * NTJ);          // 64 column-tile groups
    const int ti = wave / tilesJ;                 // 0..255
    const int tj = wave % tilesJ;                 // 0..63
    const int ibase = ti * 16;
    const int jbase = tj * 16 * NTJ;
    const int r  = lane & 15;
    const int hi = lane >> 4;

    const float* xrow = x + (size_t)(ibase + r) * DDIM + 2 * hi;

    v8f acc[NTJ];
    #pragma unroll
    for (int tgt = 0; tgt < NTJ; ++tgt) acc[tgt] = v8f{0.f,0.f,0.f,0.f,0.f,0.f,0.f,0.f};

    for (int k0 = 0; k0 < DDIM; k0 += 4) {
        v2f afrag = *(const v2f*)(xrow + k0);
        #pragma unroll
        for (int tgt = 0; tgt < NTJ; ++tgt) {
            const float* yrow = y + (size_t)(jbase + tgt * 16 + r) * DDIM + 2 * hi;
            v2f bfrag = *(const v2f*)(yrow + k0);
            acc[tgt] = __builtin_amdgcn_wmma_f32_16x16x4_f32(
                false, afrag, false, bfrag, (short)0, acc[tgt], false, false);
        }
    }

    // C/D layout: VGPR v, lane<16 -> (row ibase+v, col r); lane>=16 -> row+8
    #pragma unroll
    for (int tgt = 0; tgt < NTJ; ++tgt) {
        const int col = jbase + tgt * 16 + r;
        const float ys = ysq[col];
        #pragma unroll
        for (int v = 0; v < 8; ++v) {
            const int row = ibase + v + 8 * hi;
            P[(size_t)row * NPT + col] = IEPS * (2.f * acc[tgt][v] - xsq[row] - ys);
        }
    }
}

// -------------------- row softmin: f_i = -eps*LSE_j(log_b + g/eps + P) -----
__global__ void softmin_rows(const float* __restrict__ P, const float* __restrict__ g,
                             const float* __restrict__ log_b, float* __restrict__ fout) {
    const int lane = threadIdx.x & 31;
    const int row  = blockIdx.x * (blockDim.x >> 5) + (threadIdx.x >> 5);
    const float* Pr = P + (size_t)row * NPT;

    float m = -INFINITY;
    for (int j = lane * 4; j < NPT; j += 128) {
        float4 p  = *(const float4*)(Pr + j);
        float4 gg = *(const float4*)(g + j);
        float4 lb = *(const float4*)(log_b + j);
        float t0 = lb.x + IEPS * gg.x + p.x;
        float t1 = lb.y + IEPS * gg.y + p.y;
        float t2 = lb.z + IEPS * gg.z + p.z;
        float t3 = lb.w + IEPS * gg.w + p.w;
        m = fmaxf(m, fmaxf(fmaxf(t0, t1), fmaxf(t2, t3)));
    }
    #pragma unroll
    for (int off = 16; off > 0; off >>= 1) m = fmaxf(m, __shfl_xor(m, off, 32));

    float s = 0.f;
    for (int j = lane * 4; j < NPT; j += 128) {
        float4 p  = *(const float4*)(Pr + j);
        float4 gg = *(const float4*)(g + j);
        float4 lb = *(const float4*)(log_b + j);
        s += __expf(lb.x + IEPS * gg.x + p.x - m);
        s += __expf(lb.y + IEPS * gg.y + p.y - m);
        s += __expf(lb.z + IEPS * gg.z + p.z - m);
        s += __expf(lb.w + IEPS * gg.w + p.w - m);
    }
    #pragma unroll
    for (int off = 16; off > 0; off >>= 1) s += __shfl_xor(s, off, 32);

    if (lane == 0) fout[row] = -EPSR * (m + __logf(s));
}

// -------------------- col softmin: g_j = -eps*LSE_i(log_a + f/eps + P) -----
// Block = 8 waves owns 32 columns; wave w scans rows [w*512, w*512+512).
__global__ void softmin_cols(const float* __restrict__ P, const float* __restrict__ fv,
                             const float* __restrict__ log_a, float* __restrict__ gout) {
    __shared__ float sm[8][32];
    __shared__ float ss[8][32];
    const int lane = threadIdx.x & 31;
    const int w    = threadIdx.x >> 5;
    const int col  = blockIdx.x * 32 + lane;
    const int i0   = w * 512;

    float m = -INFINITY;
    for (int i = i0; i < i0 + 512; ++i) {
        float t = log_a[i] + IEPS * fv[i] + P[(size_t)i * NPT + col];
        m = fmaxf(m, t);
    }
    float s = 0.f;
    for (int i = i0; i < i0 + 512; ++i) {
        float t = log_a[i] + IEPS * fv[i] + P[(size_t)i * NPT + col];
        s += __expf(t - m);
    }
    sm[w][lane] = m;
    ss[w][lane] = s;
    __syncthreads();
    if (w == 0) {
        float M = -INFINITY;
        #pragma unroll
        for (int k = 0; k < 8; ++k) M = fmaxf(M, sm[k][lane]);
        float S = 0.f;
        #pragma unroll
        for (int k = 0; k < 8; ++k) S += ss[k][lane] * __expf(sm[k][lane] - M);
        gout[col] = -EPSR * (M + __logf(S));
    }
}

// -------------------- final loss: <a,f_fin> + <b,g_fin> --------------------
__global__ void loss_kernel(const float* __restrict__ a_n, const float* __restrict__ b_n,
                            const float* __restrict__ ff, const float* __restrict__ gf,
                            float* __restrict__ out) {
    __shared__ float red[1024];
    const int t = threadIdx.x;
    float s = 0.f;
    for (int i = t; i < NPT; i += 1024) s += a_n[i] * ff[i] + b_n[i] * gf[i];
    red[t] = s; __syncthreads();
    for (int off = 512; off > 0; off >>= 1) {
        if (t < off) red[t] += red[t + off];
        __syncthreads();
    }
    if (t == 0) out[0] = red[0];
}

// ---------------------------------------------------------------------------
extern "C" void kernel_launch(void* const* d_in, const int* in_sizes, int n_in,
                              void* d_out, int out_size, void* d_ws, size_t ws_size,
                              hipStream_t stream) {
    const float* x = (const float*)d_in[0];   // (4096,128)
    const float* y = (const float*)d_in[1];   // (4096,128)
    const float* a = (const float*)d_in[2];   // (4096,)
    const float* b = (const float*)d_in[3];   // (4096,)
    float* out = (float*)d_out;

    float* P = (float*)d_ws;                            // 4096*4096 f32 = 64MB
    float* vec = P + (size_t)NPT * NPT;
    float* xsq   = vec + 0 * NPT;
    float* ysq   = vec + 1 * NPT;
    float* a_n   = vec + 2 * NPT;
    float* b_n   = vec + 3 * NPT;
    float* log_a = vec + 4 * NPT;
    float* log_b = vec + 5 * NPT;
    float* f     = vec + 6 * NPT;
    float* g     = vec + 7 * NPT;
    float* f_fin = vec + 8 * NPT;
    float* g_fin = vec + 9 * NPT;

    prep_weights<<<1, 1024, 0, stream>>>(a, b, a_n, b_n, log_a, log_b, f, g);
    row_sq<<<NPT / 8, 256, 0, stream>>>(x, xsq);
    row_sq<<<NPT / 8, 256, 0, stream>>>(y, ysq);

    // 256 i-tiles * 64 j-groups = 16384 waves; 8 waves/block
    gemm_cost<<<16384 / 8, 256, 0, stream>>>(x, y, xsq, ysq, P);

    for (int it = 0; it < NIT; ++it) {                  // Gauss-Seidel
        softmin_rows<<<NPT / 8, 256, 0, stream>>>(P, g, log_b, f);
        softmin_cols<<<NPT / 32, 256, 0, stream>>>(P, f, log_a, g);
    }
    // final extrapolation: f_fin from g20, g_fin from f20 (NOT f_fin)
    softmin_rows<<<NPT / 8, 256, 0, stream>>>(P, g, log_b, f_fin);
    softmin_cols<<<NPT / 32, 256, 0, stream>>>(P, f, log_a, g_fin);

    loss_kernel<<<1, 1024, 0, stream>>>(a_n, b_n, f_fin, g_fin, out);
}